// MessageTemporalEncoding_76836964926293
// MI455X (gfx1250) — compile-verified
//
#include <hip/hip_runtime.h>
#include <math.h>

typedef float v2f __attribute__((ext_vector_type(2)));
typedef float v8f __attribute__((ext_vector_type(8)));

#define E_TOT   200000
#define DIM     512
#define NHEAD   8
#define HEADDIM 64
#define NFREQ   16
#define KF      32          // 2*NF, GEMM K dimension
#define ROWS    16          // events per block (one WMMA M tile)
#define FSTRIDE 34          // padded LDS row stride (floats), keeps 8B align

__global__ __launch_bounds__(256)
void mte_kernel(const float* __restrict__ msg,
                const float* __restrict__ t,
                const float* __restrict__ t_scale,
                const float* __restrict__ t_shift,
                const float* __restrict__ rope_log_ts,
                const float* __restrict__ fourier_freqs,
                const float* __restrict__ fourier_W,
                const float* __restrict__ fourier_b,
                const float* __restrict__ log_decay,
                const float* __restrict__ decay_bias,
                float* __restrict__ out)
{
    __shared__ float s_tn[ROWS];
    __shared__ float s_feat[ROWS * FSTRIDE];

    const int tid  = threadIdx.x;
    const int lane = tid & 31;
    const int wave = tid >> 5;          // 0..7  (== head index for phase 4)
    const int half = lane >> 4;         // 0 or 1 (wave32 half)
    const int l16  = lane & 15;
    const int e0   = blockIdx.x * ROWS;

    // Prefetch this block's msg tile (16 events * 512 f32 = 32KB; 128B/thread)
    __builtin_prefetch(msg + (size_t)e0 * DIM + tid * 32, 0, 0);

    // ---- Phase 1: t_n for the 16 events ----
    if (tid < ROWS) {
        float tv = t[e0 + tid];
        // T_MEAN=0, T_VAR=1: t_n = scale * t/(1+1e-6) + shift
        s_tn[tid] = t_scale[0] * (tv / (1.0f + 1e-6f)) + t_shift[0];
    }
    __syncthreads();

    // ---- Phase 2: fourier features feat[16][32] into LDS ----
    for (int idx = tid; idx < ROWS * KF; idx += 256) {
        int row = idx >> 5;
        int col = idx & 31;
        float phi = s_tn[row] * fourier_freqs[col & (NFREQ - 1)];
        s_feat[row * FSTRIDE + col] = (col < NFREQ) ? __sinf(phi) : __cosf(phi);
    }
    __syncthreads();

    // ---- Phase 3: fourier = feat(16x32) @ W(32x512), wave owns 64 cols ----
    const int nb = wave * 64;           // this wave's first output column
    v8f acc[4] = {};                    // 4 tiles of 16x16 f32 accum

#pragma unroll
    for (int kk = 0; kk < KF / 4; ++kk) {           // 8 K-steps of 4
        const int k0 = 4 * kk + 2 * half;           // this half-wave's K pair
        // A fragment (16x4): lane l16 -> row M=l16, holds K=k0,k0+1
        v2f a = *(const v2f*)&s_feat[l16 * FSTRIDE + k0];
#pragma unroll
        for (int tt = 0; tt < 4; ++tt) {
            const int n = nb + tt * 16 + l16;
            // B fragment (4x16): lane holds B[k0][n], B[k0+1][n]
            v2f b;
            b.x = fourier_W[(size_t)(k0 + 0) * DIM + n];
            b.y = fourier_W[(size_t)(k0 + 1) * DIM + n];
            acc[tt] = __builtin_amdgcn_wmma_f32_16x16x4_f32(
                false, a, false, b, (short)0, acc[tt], false, false);
        }
    }

    // ---- Phase 4: RoPE + gate + combine + store ----
    // All 64 columns of this wave live in head h == wave, so the decay gate
    // g(row) is tile-invariant: compute the 8 per-row gates once per lane.
    const int   h     = wave;
    const float lam   = __expf(log_decay[h]);
    const float dbias = decay_bias[h];
    float gv[8];
#pragma unroll
    for (int r = 0; r < 8; ++r) {
        const float tn = s_tn[r + 8 * half];
        gv[r] = __builtin_amdgcn_rcpf(1.0f + __expf(lam * fabsf(tn) - dbias));
    }

    // C/D layout: VGPR r, lane -> row M = r + 8*half, col N = nb+tt*16+l16
#pragma unroll
    for (int tt = 0; tt < 4; ++tt) {
        const int n    = nb + tt * 16 + l16;        // output column
        const int p    = n & (HEADDIM - 1);
        const int j    = p >> 1;                    // rope freq index
        const int odd  = p & 1;
        const float inv_ts = __expf(-rope_log_ts[h * (HEADDIM / 2) + j]);
        const float fb     = fourier_b[n];
        const int   pair   = n & ~1;

#pragma unroll
        for (int r = 0; r < 8; ++r) {
            const int row = r + 8 * half;
            const int e   = e0 + row;
            const float tn = s_tn[row];
            float s, c;
            __sincosf(tn * inv_ts, &s, &c);
            const v2f mv = *(const v2f*)(msg + (size_t)e * DIM + pair);
            const float me = mv.x, mo = mv.y;
            const float rot  = odd ? (me * s + mo * c) : (me * c - mo * s);
            const float mval = odd ? mo : me;
            const float g = gv[r];
            out[(size_t)e * DIM + n] = g * rot + (1.0f - g) * mval + acc[tt][r] + fb;
        }
    }
}

extern "C" void kernel_launch(void* const* d_in, const int* in_sizes, int n_in,
                              void* d_out, int out_size, void* d_ws, size_t ws_size,
                              hipStream_t stream) {
    const float* msg           = (const float*)d_in[0];
    const float* t             = (const float*)d_in[1];
    const float* t_scale       = (const float*)d_in[2];
    const float* t_shift       = (const float*)d_in[3];
    const float* rope_log_ts   = (const float*)d_in[4];
    const float* fourier_freqs = (const float*)d_in[5];
    const float* fourier_W     = (const float*)d_in[6];
    const float* fourier_b     = (const float*)d_in[7];
    const float* log_decay     = (const float*)d_in[8];
    const float* decay_bias    = (const float*)d_in[9];
    float* out = (float*)d_out;

    dim3 grid(E_TOT / ROWS);   // 200000/16 = 12500, exact
    dim3 block(256);           // 8 wave32 waves
    mte_kernel<<<grid, block, 0, stream>>>(msg, t, t_scale, t_shift,
                                           rope_log_ts, fourier_freqs,
                                           fourier_W, fourier_b,
                                           log_decay, decay_bias, out);
}